// WeightedFocalDiceLoss_56556129354418
// MI455X (gfx1250) — compile-verified
//
#include <hip/hip_runtime.h>
#include <cstdint>

// ---------------- problem constants ----------------
#define BATCH   4
#define NCLS    8
#define HH      256
#define WW      256
#define HWSZ    (HH * WW)                 // 65536
#define NPIX    (BATCH * HWSZ)            // 262144
#define NPLANE  (BATCH * NCLS)            // 32
#define NELEM   ((float)(BATCH * NCLS * HWSZ)) // 2097152
#define SMOOTH  1e-6f
#define INF12   1e12f

typedef __attribute__((ext_vector_type(2))) float v2f;
typedef __attribute__((ext_vector_type(8))) float v8f;

#if defined(__has_builtin)
#if __has_builtin(__builtin_amdgcn_wmma_f32_16x16x4_f32)
#define HAVE_WMMA_F32X4 1
#endif
#endif

__device__ __forceinline__ float wave_sum(float v) {
#pragma unroll
  for (int o = 16; o > 0; o >>= 1) v += __shfl_xor(v, o, 32);
  return v;
}

// ------------- accumulator layout in ws (floats) -------------
// [0] focal_sum  [1] valid_cnt  [2..9] inter[c]  [10..17] psum[c]
// [18..25] onehot_cnt[c]  [26] shape_sum
__global__ void k_init(float* gacc) {
  if (threadIdx.x < 32) gacc[threadIdx.x] = 0.0f;
}

// ------------- fused softmax + focal + dice partials -------------
__global__ void k_fused(const float* __restrict__ logits,
                        const int* __restrict__ masks,
                        float* __restrict__ probs,
                        float* __restrict__ gacc) {
  __shared__ float sAcc[26];
  const int tid = threadIdx.x;
  if (tid < 26) sAcc[tid] = 0.0f;
  __syncthreads();

  const int pix = blockIdx.x * 256 + tid;       // grid covers NPIX exactly
  const int b   = pix >> 16;                    // / HWSZ
  const int hw  = pix & (HWSZ - 1);
  const size_t base = (size_t)b * NCLS * HWSZ + hw;

  float l[8];
#pragma unroll
  for (int c = 0; c < NCLS; ++c) l[c] = logits[base + (size_t)c * HWSZ];

  float m = l[0];
#pragma unroll
  for (int c = 1; c < NCLS; ++c) m = fmaxf(m, l[c]);

  float e[8], s = 0.0f;
#pragma unroll
  for (int c = 0; c < NCLS; ++c) { e[c] = expf(l[c] - m); s += e[c]; }
  const float inv = 1.0f / s;

#pragma unroll
  for (int c = 0; c < NCLS; ++c) probs[base + (size_t)c * HWSZ] = e[c] * inv;

  const int mk    = masks[pix];
  const int valid = (mk != 255);
  const int t     = (valid ? mk : 0) & 7;

  float lt = 0.0f, pt = 0.0f;
#pragma unroll
  for (int c = 0; c < NCLS; ++c) {
    lt += (t == c) ? l[c] : 0.0f;
    pt += (t == c) ? e[c] * inv : 0.0f;
  }
  const float logp_t = lt - m - logf(s);
  const float omp    = 1.0f - pt;
  const float fl     = valid ? (omp * omp * (-logp_t)) : 0.0f;

  const int lane = tid & 31;
  float r0 = wave_sum(fl);
  float r1 = wave_sum(valid ? 1.0f : 0.0f);
  if (lane == 0) { atomicAdd(&sAcc[0], r0); atomicAdd(&sAcc[1], r1); }

#pragma unroll
  for (int c = 0; c < NCLS; ++c) {
    const float pc = e[c] * inv;
    float ri = wave_sum((t == c) ? pc : 0.0f);
    float rp = wave_sum(pc);
    float ro = wave_sum((t == c) ? 1.0f : 0.0f);
    if (lane == 0) {
      atomicAdd(&sAcc[2 + c], ri);
      atomicAdd(&sAcc[10 + c], rp);
      atomicAdd(&sAcc[18 + c], ro);
    }
  }
  __syncthreads();
  if (tid < 26) atomicAdd(&gacc[tid], sAcc[tid]);
}

// ------------- EDT column pass: exact 1D binary DT (two scans) -------------
// G[p][i][j] <- squared column distance (or 1e12 if class absent in column)
__global__ void k_col(const int* __restrict__ masks, float* __restrict__ G) {
  const int idx = blockIdx.x * 256 + threadIdx.x;    // [0, NPLANE*WW)
  const int j = idx & (WW - 1);
  const int p = idx >> 8;                            // plane = b*8 + c
  const int b = p >> 3;
  const int c = p & 7;
  const int* mcol   = masks + (size_t)b * HWSZ + j;
  float*     gcol   = G + (size_t)p * HWSZ + j;
  const int  BIG    = 1 << 20;

  int d = BIG;
  for (int i = 0; i < HH; ++i) {
    d = (mcol[i * WW] == c) ? 0 : (d + 1);
    gcol[i * WW] = (float)d;
  }
  d = BIG;
  for (int i = HH - 1; i >= 0; --i) {
    d = (mcol[i * WW] == c) ? 0 : (d + 1);
    float md = fminf(gcol[i * WW], (float)d);
    gcol[i * WW] = (md >= 524288.0f) ? INF12 : md * md;
  }
}

// ------------- EDT row pass + shape-loss accumulation -------------
// One block per (plane,row). Row of g2 staged to LDS via CDNA5 async-to-LDS.
// Outward scan with early exit: once r^2 >= best, no k can improve the min.
// Block partial sums reduced on the matrix pipe via v_wmma_f32_16x16x4_f32.
__global__ void k_row(const float* __restrict__ G,
                      const float* __restrict__ probs,
                      float* __restrict__ gacc) {
  __shared__ float g2[WW];
  __shared__ float red[WW];
  const int tid = threadIdx.x;           // = j
  const int row = blockIdx.x;            // [0, NPLANE*HH)
  const int p   = row >> 8;
  const int i   = row & (HH - 1);
  const size_t rbase = ((size_t)p * HH + i) * WW;

  // --- async stage one row (1 KB) of squared column distances into LDS ---
  {
    unsigned laddr = (unsigned)(uintptr_t)(&g2[tid]);
    unsigned long long gaddr = (unsigned long long)(uintptr_t)(&G[rbase + tid]);
    asm volatile("global_load_async_to_lds_b32 %0, %1, off"
                 :: "v"(laddr), "v"(gaddr) : "memory");
    asm volatile("s_wait_asynccnt 0x0" ::: "memory");
  }
  __syncthreads();

  // --- exact min-plus with early exit ---
  float best = g2[tid];                  // k == j term (dj = 0)
  for (int r = 1; r < WW; ++r) {
    const float rr = (float)(r * r);
    if (rr >= best) break;
    const int kl = tid - r, kr = tid + r;
    if (kl >= 0) best = fminf(best, g2[kl] + rr);
    if (kr < WW) best = fminf(best, g2[kr] + rr);
  }
  const float val = sqrtf(best) * probs[rbase + tid];

  // --- block reduction: matrix-pipe (WMMA) fold of 256 partials ---
  red[tid] = val;
  __syncthreads();
  if (tid < 32) {
#if defined(HAVE_WMMA_F32X4)
    v8f acc = {};
    v2f ones; ones.x = 1.0f; ones.y = 1.0f;   // A = all-ones 16x4
#pragma unroll
    for (int t = 0; t < 4; ++t) {
      v2f bm;                                 // B carries 64 partials
      bm.x = red[tid + 64 * t];
      bm.y = red[tid + 32 + 64 * t];
      // D[m,n] = sum_k B[k,n] + C[m,n]  (column sums, replicated over 16 rows)
      acc = __builtin_amdgcn_wmma_f32_16x16x4_f32(
          false, ones, false, bm, (short)0, acc, false, false);
    }
    float s = acc[0] + acc[1] + acc[2] + acc[3] +
              acc[4] + acc[5] + acc[6] + acc[7];
    s = wave_sum(s) * (1.0f / 16.0f);         // undo 16x row replication
#else
    float s = 0.0f;
#pragma unroll
    for (int t = 0; t < 8; ++t) s += red[tid + 32 * t];
    s = wave_sum(s);
#endif
    if (tid == 0) atomicAdd(&gacc[26], s);
  }
}

// ------------- finalize scalars -------------
__global__ void k_fin(const float* __restrict__ gacc, float* __restrict__ out) {
  if (threadIdx.x != 0 || blockIdx.x != 0) return;
  const float nv      = fmaxf(gacc[1], 1.0f);
  const float focal_l = gacc[0] / nv;
  float acc = 0.0f;
#pragma unroll
  for (int c = 0; c < NCLS; ++c) {
    const float inter = gacc[2 + c];
    const float card  = gacc[10 + c] + gacc[18 + c];
    acc += (2.0f * inter + SMOOTH) / (card + SMOOTH);
  }
  const float dice_l  = 1.0f - acc / (float)NCLS;
  const float shape_l = gacc[26] / NELEM;
  out[0] = focal_l + dice_l + shape_l;
  out[1] = focal_l;
  out[2] = dice_l;
  out[3] = shape_l;
}

extern "C" void kernel_launch(void* const* d_in, const int* in_sizes, int n_in,
                              void* d_out, int out_size, void* d_ws, size_t ws_size,
                              hipStream_t stream) {
  const float* logits = (const float*)d_in[0];
  const int*   masks  = (const int*)d_in[1];
  float*       out    = (float*)d_out;

  float* wsf   = (float*)d_ws;
  float* gacc  = wsf;                       // 32 floats
  float* probs = wsf + 64;                  // B*C*H*W floats (8 MB)
  float* G     = probs + (size_t)BATCH * NCLS * HWSZ; // 8 MB

  k_init <<<1, 32, 0, stream>>>(gacc);
  k_fused<<<NPIX / 256, 256, 0, stream>>>(logits, masks, probs, gacc);
  k_col  <<<(NPLANE * WW) / 256, 256, 0, stream>>>(masks, G);
  k_row  <<<NPLANE * HH, 256, 0, stream>>>(G, probs, gacc);
  k_fin  <<<1, 32, 0, stream>>>(gacc, out);
}